// ReLie_10685878633092
// MI455X (gfx1250) — compile-verified
//
#include <hip/hip_runtime.h>
#include <stdint.h>

#define N_MC    32
#define M_TOT   8192
#define D_DIM   16
#define KMAX    5
#define KTERMS  11
#define TILE_M  32         // m-columns per block; lane == m_local (wave32)
#define PAD     17         // padded row stride (floats) -> conflict-free for lane=m
#define N_OUTER 2          // n-chunks per thread; gridDim.y covers the rest

typedef __attribute__((ext_vector_type(4))) float v4f;   // clang vector: OK for nontemporal builtins

typedef __attribute__((address_space(3))) const void* lds_cptr_t;

__device__ __forceinline__ uint32_t lds_off_u32(const void* p) {
    // generic -> LDS addrspace cast; yields the 32-bit LDS byte address
    return (uint32_t)(uintptr_t)(lds_cptr_t)p;
}

__global__ __launch_bounds__(256)
void relie_wrapped_normal_kernel(const float* __restrict__ eps,
                                 const float* __restrict__ gamma_raw,
                                 const float* __restrict__ mu,
                                 float* __restrict__ out_g,
                                 float* __restrict__ out_lq)
{
    __shared__ __align__(16) float raw_g[TILE_M * D_DIM];   // 2 KB staged async
    __shared__ __align__(16) float raw_m[TILE_M * D_DIM];   // 2 KB staged async
    __shared__ float sh_scale [TILE_M * PAD];
    __shared__ float sh_iscale[TILE_M * PAD];
    __shared__ float sh_lnrm  [TILE_M * PAD];
    __shared__ float sh_mu    [TILE_M * PAD];

    const int tid  = threadIdx.x;
    const int lane = tid & 31;
    const int wave = tid >> 5;
    const int m0   = blockIdx.x * TILE_M;

    // ---- CDNA5 async DMA: stage gamma_raw / mu tiles into LDS (ASYNCcnt path) ----
    {
        const int      chunk = (wave & 3) * 32 + lane;        // 0..127 (16B chunks of 2KB)
        const uint32_t voff  = (uint32_t)chunk * 16u;         // per-lane byte offset
        if (wave < 4) {
            const uint64_t src = (uint64_t)(uintptr_t)(gamma_raw + (size_t)m0 * D_DIM);
            const uint32_t dst = lds_off_u32(raw_g) + voff;
            asm volatile("global_load_async_to_lds_b128 %0, %1, %2"
                         :: "v"(dst), "v"(voff), "s"(src) : "memory");
        } else {
            const uint64_t src = (uint64_t)(uintptr_t)(mu + (size_t)m0 * D_DIM);
            const uint32_t dst = lds_off_u32(raw_m) + voff;
            asm volatile("global_load_async_to_lds_b128 %0, %1, %2"
                         :: "v"(dst), "v"(voff), "s"(src) : "memory");
        }
        asm volatile("s_wait_asynccnt 0" ::: "memory");
    }
    __syncthreads();

    // ---- derive per-(m,d) params once; reused by the MC samples of this block ----
    #pragma unroll
    for (int p = tid; p < TILE_M * D_DIM; p += 256) {
        const int mm = p >> 4, dd = p & 15;
        const float gr = raw_g[p];
        // softplus, numerically robust
        const float s  = fmaxf(gr, 0.0f) + log1pf(__expf(-fabsf(gr)));
        const int   o  = mm * PAD + dd;
        sh_scale [o] = s;
        sh_iscale[o] = 1.0f / s;
        sh_lnrm  [o] = -0.91893853320467274178f - __logf(s);  // -0.5*log(2pi) - log(s)
        sh_mu    [o] = raw_m[p];
    }
    __syncthreads();

    const int   m          = m0 + lane;
    const int   n_base     = blockIdx.y * (8 * N_OUTER) + wave;   // gridDim.y = 32/(8*N_OUTER)
    const float TWO_PI_F   = 6.28318530717958647692f;
    const float INV_TWO_PI = 0.15915494309189533577f;

    float lq[N_OUTER];
    #pragma unroll
    for (int i = 0; i < N_OUTER; ++i) lq[i] = 0.0f;

    #pragma unroll
    for (int q = 0; q < 4; ++q) {                       // d-groups of 4
        float sc[4], is[4], ln[4], mv[4];
        #pragma unroll
        for (int j = 0; j < 4; ++j) {
            const int o = lane * PAD + (q * 4 + j);
            sc[j] = sh_scale[o];  is[j] = sh_iscale[o];
            ln[j] = sh_lnrm[o];   mv[j] = sh_mu[o];
        }
        #pragma unroll
        for (int i = 0; i < N_OUTER; ++i) {             // n = n_base + 8*i
            const int    n    = n_base + 8 * i;
            const size_t base = ((size_t)n * M_TOT + m) * D_DIM;
            if (q == 0 && i + 1 < N_OUTER)              // prefetch next n-chunk line
                __builtin_prefetch(eps + (((size_t)(n + 8) * M_TOT + m) * D_DIM), 0, 1);
            // streaming read-once data: non-temporal (TH_LOAD_NT)
            const v4f ev = __builtin_nontemporal_load((const v4f*)(eps + base + q * 4));
            const float ex[4] = {ev.x, ev.y, ev.z, ev.w};
            float gx[4];
            #pragma unroll
            for (int j = 0; j < 4; ++j) {
                const float x = ex[j] * sc[j];
                float t[KTERMS];
                float mx = -3.4e38f;
                #pragma unroll
                for (int k = 0; k < KTERMS; ++k) {
                    const float z = (x + TWO_PI_F * (float)(k - KMAX)) * is[j];
                    t[k] = -0.5f * z * z;
                    mx = fmaxf(mx, t[k]);
                }
                float ssum = 0.0f;
                #pragma unroll
                for (int k = 0; k < KTERMS; ++k)
                    ssum += __expf(t[k] - mx);
                lq[i] += ln[j] + mx + __logf(ssum);
                float r = x + mv[j];
                r = r - TWO_PI_F * floorf(r * INV_TWO_PI);   // mod 2pi, result in [0,2pi)
                gx[j] = r;
            }
            v4f gv; gv.x = gx[0]; gv.y = gx[1]; gv.z = gx[2]; gv.w = gx[3];
            // write-once data: non-temporal store (TH_STORE_NT)
            __builtin_nontemporal_store(gv, (v4f*)(out_g + base + q * 4));
        }
    }

    #pragma unroll
    for (int i = 0; i < N_OUTER; ++i) {
        const int n = n_base + 8 * i;
        __builtin_nontemporal_store(lq[i], out_lq + (size_t)n * M_TOT + m);
    }
}

extern "C" void kernel_launch(void* const* d_in, const int* in_sizes, int n_in,
                              void* d_out, int out_size, void* d_ws, size_t ws_size,
                              hipStream_t stream) {
    const float* eps       = (const float*)d_in[0];
    const float* gamma_raw = (const float*)d_in[1];
    const float* mu        = (const float*)d_in[2];
    float* out_g  = (float*)d_out;
    float* out_lq = (float*)d_out + (size_t)N_MC * M_TOT * D_DIM;

    dim3 grid(M_TOT / TILE_M, N_MC / (8 * N_OUTER));   // 256 x 2 = 512 blocks
    dim3 block(256);                                   // 8 wave32s
    relie_wrapped_normal_kernel<<<grid, block, 0, stream>>>(eps, gamma_raw, mu, out_g, out_lq);
}